// MeshRendererShadow_36197984371118
// MI455X (gfx1250) — compile-verified
//
#include <hip/hip_runtime.h>

// PCF soft-shadow visibility for MI455X (gfx1250, wave32).
// Memory-bound gather kernel: zbuf (128MB) is L2-resident (192MB L2);
// streaming arrays use non-temporal hints so they don't evict it.

typedef int vint2 __attribute__((ext_vector_type(2)));

#define PCF_BIAS 0.008f
// SHARPNESS * log2(e): sigmoid(t) = 1/(1+exp2(-t*log2e))
#define PCF_K (1000.0f * 1.44269504088896340736f)

__global__ __launch_bounds__(256) void pcf_shadow_kernel(
    const float* __restrict__ zbuf,    // [N, S, S]
    const float* __restrict__ depth,   // [N, H, W] (K=1)
    const vint2* __restrict__ xy,      // [N, H, W] pairs (x, y)
    const int*   __restrict__ imsize,  // scalar S on device
    float* __restrict__ out,           // [N, H, W]
    int zbuf_count,                    // N*S*S
    int total)                         // N*H*W
{
    const int idx = blockIdx.x * (int)blockDim.x + threadIdx.x;
    if (idx >= total) return;

    // Uniform shape math (all SGPR): S from device scalar, N and HW derived.
    const int S  = *imsize;
    const int SS = S * S;
    const int N  = zbuf_count / SS;    // uniform scalar division
    const int HW = total / N;          // uniform scalar division

    const int n = idx / HW;            // batch index (division by uniform)

    // Streaming loads: non-temporal so the L2 keeps the z-buffer resident.
    vint2 p = __builtin_nontemporal_load(xy + idx);
    const int x = p.x;
    const int y = p.y;
    const float aa = __builtin_nontemporal_load(depth + idx) - PCF_BIAS;

    const float* plane = zbuf + (long)n * (long)SS;

    // Contiguous 3-float x-window covering all clamped taps:
    //   taps are clip(x-1), clip(x), clip(x+1), all inside [bx, bx+2].
    const int bx   = min(max(x - 1, 0), S - 3);
    const int off0 = min(max(x - 1, 0), S - 1) - bx;  // 0 or 1
    const int off1 = x - bx;                          // 0, 1, or 2
    const int off2 = min(x + 1, S - 1) - bx;          // 1 or 2

    float acc = 0.0f;
    #pragma unroll
    for (int ii = -1; ii <= 1; ++ii) {
        const int yi = min(max(y + ii, 0), S - 1);
        const float* row = plane + (long)yi * (long)S + bx;
        // 3 adjacent dword loads -> merged into one global_load_b96 gather.
        const float v0 = row[0];
        const float v1 = row[1];
        const float v2 = row[2];
        // Branch-free clamped-tap selection (v_cndmask).
        const float t0 = off0 ? v1 : v0;
        const float t1 = (off1 == 1) ? v1 : ((off1 == 0) ? v0 : v2);
        const float t2 = (off2 == 1) ? v1 : v2;
        // sigmoid((bb - aa) * 1000) via v_exp_f32 (exp2) + v_rcp_f32.
        acc += __builtin_amdgcn_rcpf(1.0f + __builtin_amdgcn_exp2f((aa - t0) * PCF_K));
        acc += __builtin_amdgcn_rcpf(1.0f + __builtin_amdgcn_exp2f((aa - t1) * PCF_K));
        acc += __builtin_amdgcn_rcpf(1.0f + __builtin_amdgcn_exp2f((aa - t2) * PCF_K));
    }

    __builtin_nontemporal_store(acc * (1.0f / 9.0f), out + idx);
}

extern "C" void kernel_launch(void* const* d_in, const int* in_sizes, int n_in,
                              void* d_out, int out_size, void* d_ws, size_t ws_size,
                              hipStream_t stream) {
    const float* zbuf   = (const float*)d_in[0];  // light_zbuf [N,S,S] f32
    const float* depth  = (const float*)d_in[1];  // depth_z [N,H,W,1] f32
    const vint2* xy     = (const vint2*)d_in[2];  // xy [N,H,W,1,2] i32
    const int*   imsize = (const int*)d_in[3];    // scalar S
    float* out = (float*)d_out;

    const int zbuf_count = in_sizes[0];  // N*S*S
    const int total      = in_sizes[1];  // N*H*W*K
    const int threads = 256;             // 8 wave32s per block
    const int blocks  = (total + threads - 1) / threads;

    pcf_shadow_kernel<<<blocks, threads, 0, stream>>>(
        zbuf, depth, xy, imsize, out, zbuf_count, total);
}